// SplitCrossEntropyLoss_53386443489982
// MI455X (gfx1250) — compile-verified
//
#include <hip/hip_runtime.h>
#include <hip/hip_bf16.h>
#include <math.h>

// ---------------------------------------------------------------------------
// Split (adaptive) softmax cross-entropy for MI455X (gfx1250, wave32).
// - fused online-LSE GEMMs, V_WMMA_F32_16X16X4_F32 (fp32-exact vs reference)
// - 64 rows per block (FOUR C tiles per wave): 4x B reuse (0.25 global b64
//   per WMMA), 4 independent WMMA chains; 263KB LDS A-tile (CDNA5 320KB/WGP)
// - all three splits + 4-way column split in ONE launch, deterministic
//   (m,s) partials merged by a follow-up kernel (no float atomics)
// ---------------------------------------------------------------------------

typedef __attribute__((ext_vector_type(2))) float v2f;
typedef __attribute__((ext_vector_type(8))) float v8f;

#define HID 1024
#define NROWS 4096
#define ASTRIDE 1028   // 1024 + 4 pad: float4-aligned, spreads LDS banks
#define ROWT 64        // rows per block (four 16-row WMMA tiles)
#define NCHUNK 4       // column chunks per row tile
#define RTMAX 64       // NROWS / ROWT
#define BLKS_PER_PASS (RTMAX * NCHUNK)   // 256

__device__ __forceinline__ float half_max(float x) {
  x = fmaxf(x, __shfl_xor(x, 1, 32));
  x = fmaxf(x, __shfl_xor(x, 2, 32));
  x = fmaxf(x, __shfl_xor(x, 4, 32));
  x = fmaxf(x, __shfl_xor(x, 8, 32));
  return x;
}
__device__ __forceinline__ float half_sum(float x) {
  x += __shfl_xor(x, 1, 32);
  x += __shfl_xor(x, 2, 32);
  x += __shfl_xor(x, 4, 32);
  x += __shfl_xor(x, 8, 32);
  return x;
}

// ---------------------------------------------------------------------------
// Kernel 0: bucket assignment + deterministic compaction of bucket-1/2 rows.
// ---------------------------------------------------------------------------
__global__ __launch_bounds__(256)
void bucket_compact(const int* __restrict__ targets,
                    int* __restrict__ bucket,
                    int* __restrict__ list1,
                    int* __restrict__ list2,
                    int* __restrict__ counts) {
  __shared__ int c1[256], c2[256];
  const int t = threadIdx.x;
  const int base = t * 16;
  int l1 = 0, l2 = 0;
#pragma unroll
  for (int i = 0; i < 16; ++i) {
    int tg = targets[base + i];
    int b = (tg >= 4000 ? 1 : 0) + (tg >= 20000 ? 1 : 0);
    bucket[base + i] = b;
    l1 += (b == 1);
    l2 += (b == 2);
  }
  c1[t] = l1; c2[t] = l2;
  __syncthreads();
  if (t == 0) {                 // serial exclusive scan (deterministic)
    int r1 = 0, r2 = 0;
    for (int j = 0; j < 256; ++j) {
      int a = c1[j], b = c2[j];
      c1[j] = r1; c2[j] = r2;
      r1 += a; r2 += b;
    }
    counts[0] = r1;
    counts[1] = r2;
  }
  __syncthreads();
  int p1 = c1[t], p2 = c2[t];
  for (int i = 0; i < 16; ++i) {
    int b = bucket[base + i];
    if (b == 1) list1[p1++] = base + i;
    else if (b == 2) list2[p2++] = base + i;
  }
}

// ---------------------------------------------------------------------------
// One launch covers all three splits.
//   blockIdx: pass = idx/256; rt = (idx%256)/4; chunk = idx%4
//   wave column stripe: ct = chunk*8 + wave, step 32
// Each wave: four 16x16 C tiles (rows nlo+{0,16,32,48} of the 64-row A tile)
// sharing each B load. Per-block (m, s) partials per slot/chunk -> merge.
// ---------------------------------------------------------------------------
__global__ __launch_bounds__(256)
void lse_gemm_all(const float* __restrict__ h,
                  const float* __restrict__ weight,
                  const float* __restrict__ bias,
                  const int* __restrict__ list1,
                  const int* __restrict__ list2,
                  const int* __restrict__ counts,
                  float2* __restrict__ P) {  // [3][NROWS][NCHUNK] (m, s)
  __shared__ float As[ROWT * ASTRIDE];       // 263 KB (CDNA5 LDS)
  __shared__ float red[8][ROWT][2];

  const int pass  = blockIdx.x >> 8;        // 0=head 1=tail1 2=tail2
  const int rem   = blockIdx.x & 255;
  const int rt    = rem >> 2;
  const int chunk = rem & 3;

  const int colstart = (pass == 0) ? 0 : (pass == 1 ? 4000 : 20000);
  const int ntiles   = (pass == 0) ? 250 : (pass == 1 ? 1000 : 1875);
  const int count    = (pass == 0) ? NROWS : counts[pass - 1];
  const int* rowlist = (pass == 0) ? (const int*)nullptr
                                   : (pass == 1 ? list1 : list2);
  if (rt * ROWT >= count) return;

  const float* W = weight + (long)colstart * HID;
  const float* B = bias + colstart;

  const int t = threadIdx.x;
  // --- stage 64 x 1024 fp32 A tile into LDS (coalesced float4) ---
#pragma unroll 4
  for (int i = 0; i < ROWT; ++i) {
    int slot = rt * ROWT + i;
    int srow = slot < count ? slot : count - 1;
    int row = rowlist ? rowlist[srow] : srow;
    float4 v = ((const float4*)(h + (long)row * HID))[t];
    *(float4*)&As[i * ASTRIDE + t * 4] = v;
  }
  __syncthreads();

  const int wave = t >> 5;
  const int lane = t & 31;
  const int nlo = lane & 15;
  const int hi = lane >> 4;

  float m0[8], s0[8], m1[8], s1[8], m2[8], s2[8], m3[8], s3[8];
#pragma unroll
  for (int v = 0; v < 8; ++v) {
    m0[v] = -INFINITY; s0[v] = 0.0f;
    m1[v] = -INFINITY; s1[v] = 0.0f;
    m2[v] = -INFINITY; s2[v] = 0.0f;
    m3[v] = -INFINITY; s3[v] = 0.0f;
  }

  const float* a0row = As + (nlo +  0) * ASTRIDE + 2 * hi;
  const float* a1row = As + (nlo + 16) * ASTRIDE + 2 * hi;
  const float* a2row = As + (nlo + 32) * ASTRIDE + 2 * hi;
  const float* a3row = As + (nlo + 48) * ASTRIDE + 2 * hi;

  for (int ct = chunk * 8 + wave; ct < ntiles; ct += 32) {
    const int col = ct * 16 + nlo;
    const float* wrow = W + (long)col * HID + 2 * hi;

    if (ct + 32 < ntiles)    // warm this wave's next column tile
      __builtin_prefetch(W + (long)((ct + 32) * 16 + nlo) * HID, 0, 1);

    v8f c0 = {};
    v8f c1 = {};
    v8f c2 = {};
    v8f c3 = {};
#pragma unroll 4
    for (int k = 0; k < HID; k += 4) {
      v2f b  = *(const v2f*)(wrow + k);    // one global_load_b64 ...
      v2f a0 = *(const v2f*)(a0row + k);   // ... feeds four WMMAs
      v2f a1 = *(const v2f*)(a1row + k);
      v2f a2 = *(const v2f*)(a2row + k);
      v2f a3 = *(const v2f*)(a3row + k);
      c0 = __builtin_amdgcn_wmma_f32_16x16x4_f32(
          false, a0, false, b, (short)0, c0, false, false);
      c1 = __builtin_amdgcn_wmma_f32_16x16x4_f32(
          false, a1, false, b, (short)0, c1, false, false);
      c2 = __builtin_amdgcn_wmma_f32_16x16x4_f32(
          false, a2, false, b, (short)0, c2, false, false);
      c3 = __builtin_amdgcn_wmma_f32_16x16x4_f32(
          false, a3, false, b, (short)0, c3, false, false);
    }

    const float bcol = B[col];
#pragma unroll
    for (int v = 0; v < 8; ++v) {
      {
        float x = c0[v] + bcol;
        float nm = fmaxf(m0[v], half_max(x));
        float e = half_sum(__expf(x - nm));
        s0[v] = s0[v] * __expf(m0[v] - nm) + e;
        m0[v] = nm;
      }
      {
        float x = c1[v] + bcol;
        float nm = fmaxf(m1[v], half_max(x));
        float e = half_sum(__expf(x - nm));
        s1[v] = s1[v] * __expf(m1[v] - nm) + e;
        m1[v] = nm;
      }
      {
        float x = c2[v] + bcol;
        float nm = fmaxf(m2[v], half_max(x));
        float e = half_sum(__expf(x - nm));
        s2[v] = s2[v] * __expf(m2[v] - nm) + e;
        m2[v] = nm;
      }
      {
        float x = c3[v] + bcol;
        float nm = fmaxf(m3[v], half_max(x));
        float e = half_sum(__expf(x - nm));
        s3[v] = s3[v] * __expf(m3[v] - nm) + e;
        m3[v] = nm;
      }
    }
  }

  // --- merge the 8 waves' partials for the 64 rows ---
  if (nlo == 0) {
#pragma unroll
    for (int v = 0; v < 8; ++v) {
      red[wave][ 0 + v + 8 * hi][0] = m0[v];
      red[wave][ 0 + v + 8 * hi][1] = s0[v];
      red[wave][16 + v + 8 * hi][0] = m1[v];
      red[wave][16 + v + 8 * hi][1] = s1[v];
      red[wave][32 + v + 8 * hi][0] = m2[v];
      red[wave][32 + v + 8 * hi][1] = s2[v];
      red[wave][48 + v + 8 * hi][0] = m3[v];
      red[wave][48 + v + 8 * hi][1] = s3[v];
    }
  }
  __syncthreads();
  if (t < ROWT) {
    float M = -INFINITY;
#pragma unroll
    for (int w = 0; w < 8; ++w) M = fmaxf(M, red[w][t][0]);
    float S = 0.0f;
#pragma unroll
    for (int w = 0; w < 8; ++w) S += red[w][t][1] * __expf(red[w][t][0] - M);
    int slot = rt * ROWT + t;
    P[((long)pass * NROWS + slot) * NCHUNK + chunk] = make_float2(M, S);
  }
}

// ---------------------------------------------------------------------------
// Merge the NCHUNK column-split partials into per-row LSE values.
// ---------------------------------------------------------------------------
__device__ __forceinline__ float merge4(const float2* p) {
  float M = -INFINITY;
#pragma unroll
  for (int c = 0; c < NCHUNK; ++c) M = fmaxf(M, p[c].x);
  float S = 0.0f;
#pragma unroll
  for (int c = 0; c < NCHUNK; ++c) S += p[c].y * __expf(p[c].x - M);
  return M + __logf(S);
}

__global__ __launch_bounds__(256)
void merge_lse(const float2* __restrict__ P,
               const int* __restrict__ counts,
               const int* __restrict__ list1,
               const int* __restrict__ list2,
               float* __restrict__ lse_head,
               float* __restrict__ lse_tail) {
  const int idx = blockIdx.x * 256 + threadIdx.x;   // 0..4095
  lse_head[idx] = merge4(P + (long)idx * NCHUNK);
  if (idx < counts[0])
    lse_tail[list1[idx]] = merge4(P + ((long)NROWS + idx) * NCHUNK);
  if (idx < counts[1])
    lse_tail[list2[idx]] = merge4(P + ((long)2 * NROWS + idx) * NCHUNK);
}

// ---------------------------------------------------------------------------
// Per-row finalize: 3 dot products (target weight row, two tail vectors),
// fold tail-vector logits into head LSE, form loss; one wave per row.
// ---------------------------------------------------------------------------
__global__ __launch_bounds__(256)
void finalize_rows(const float* __restrict__ h,
                   const float* __restrict__ weight,
                   const float* __restrict__ bias,
                   const float* __restrict__ tail_vectors,
                   const float* __restrict__ tail_bias,
                   const int*   __restrict__ targets,
                   const int*   __restrict__ bucket,
                   const float* __restrict__ lse_head,
                   const float* __restrict__ lse_tail,
                   float* __restrict__ partial) {
  __shared__ float sh[8];
  const int wave = threadIdx.x >> 5;
  const int lane = threadIdx.x & 31;
  const int r = blockIdx.x * 8 + wave;

  const int tgt = targets[r];
  const float4* hv = (const float4*)(h + (long)r * HID);
  const float4* wt = (const float4*)(weight + (long)tgt * HID);
  const float4* t0 = (const float4*)(tail_vectors);
  const float4* t1 = (const float4*)(tail_vectors + HID);

  float d = 0.0f, e0 = 0.0f, e1 = 0.0f;
#pragma unroll
  for (int i = 0; i < 8; ++i) {
    float4 a = hv[lane + i * 32];
    float4 b = wt[lane + i * 32];
    float4 c0 = t0[lane + i * 32];
    float4 c1 = t1[lane + i * 32];
    d  += a.x * b.x  + a.y * b.y  + a.z * b.z  + a.w * b.w;
    e0 += a.x * c0.x + a.y * c0.y + a.z * c0.z + a.w * c0.w;
    e1 += a.x * c1.x + a.y * c1.y + a.z * c1.z + a.w * c1.w;
  }
#pragma unroll
  for (int msk = 16; msk >= 1; msk >>= 1) {
    d  += __shfl_xor(d,  msk, 32);
    e0 += __shfl_xor(e0, msk, 32);
    e1 += __shfl_xor(e1, msk, 32);
  }

  if (lane == 0) {
    d  += bias[tgt];
    e0 += tail_bias[0];
    e1 += tail_bias[1];
    float L4 = lse_head[r];
    float mx = fmaxf(L4, fmaxf(e0, e1));
    float LH = mx + __logf(__expf(L4 - mx) + __expf(e0 - mx) + __expf(e1 - mx));
    int b = bucket[r];
    float loss;
    if (b == 0) {
      loss = -(d - LH);
    } else {
      float head_col = (b == 1) ? e1 : e0;   // head cols 4001 / 4000
      loss = -((head_col - LH) + (d - lse_tail[r]));
    }
    sh[wave] = loss;
  }
  __syncthreads();
  if (threadIdx.x == 0) {
    float sum = 0.0f;
#pragma unroll
    for (int w = 0; w < 8; ++w) sum += sh[w];
    partial[blockIdx.x] = sum;
  }
}

__global__ __launch_bounds__(512)
void reduce_out(const float* __restrict__ partial, float* __restrict__ out) {
  __shared__ float sm[512];
  const int t = threadIdx.x;
  sm[t] = partial[t];
  __syncthreads();
  for (int ofs = 256; ofs > 0; ofs >>= 1) {
    if (t < ofs) sm[t] += sm[t + ofs];
    __syncthreads();
  }
  if (t == 0) out[0] = sm[0] * (1.0f / (float)NROWS);
}

// ---------------------------------------------------------------------------
extern "C" void kernel_launch(void* const* d_in, const int* in_sizes, int n_in,
                              void* d_out, int out_size, void* d_ws, size_t ws_size,
                              hipStream_t stream) {
  const float* weight       = (const float*)d_in[0];   // 50000 x 1024
  const float* bias         = (const float*)d_in[1];   // 50000
  const float* hiddens      = (const float*)d_in[2];   // 8 x 512 x 1024
  const int*   targets      = (const int*)  d_in[3];   // 4096
  const float* tail_vectors = (const float*)d_in[4];   // 2 x 1024
  const float* tail_bias    = (const float*)d_in[5];   // 2

  char* ws = (char*)d_ws;
  float*  lse_head = (float*) (ws + 0);                // 4096 f32
  float*  lse_tail = (float*) (ws + 16 * 1024);        // 4096 f32
  int*    bucket   = (int*)   (ws + 32 * 1024);        // 4096 i32
  int*    list1    = (int*)   (ws + 48 * 1024);        // 4096 i32
  int*    list2    = (int*)   (ws + 64 * 1024);        // 4096 i32
  int*    counts   = (int*)   (ws + 80 * 1024);        // 2 i32
  float*  partial  = (float*) (ws + 84 * 1024);        // 512 f32
  float2* P        = (float2*)(ws + 96 * 1024);        // 3*4096*4 float2 = 384KB

  bucket_compact<<<1, 256, 0, stream>>>(targets, bucket, list1, list2, counts);

  lse_gemm_all<<<3 * BLKS_PER_PASS, 256, 0, stream>>>(
      hiddens, weight, bias, list1, list2, counts, P);

  merge_lse<<<16, 256, 0, stream>>>(P, counts, list1, list2, lse_head, lse_tail);

  finalize_rows<<<512, 256, 0, stream>>>(hiddens, weight, bias,
                                         tail_vectors, tail_bias, targets, bucket,
                                         lse_head, lse_tail, partial);
  reduce_out<<<1, 512, 0, stream>>>(partial, (float*)d_out);
}